// SAGEModel_42528766165365
// MI455X (gfx1250) — compile-verified
//
#include <hip/hip_runtime.h>
#include <hip/hip_fp16.h>

typedef __attribute__((ext_vector_type(2)))  float v2f;
typedef __attribute__((ext_vector_type(8)))  float v8f;

#define NN 100000    // nodes
#define NE 1600000   // edges (self loops handled separately)
#define DD 128       // feature dim
#define HH 256       // mlp hidden
#define LL 40        // labels

// ---------------- degree / normalization ----------------

__global__ void k_init_deg(float* __restrict__ deg) {
  int i = blockIdx.x * blockDim.x + threadIdx.x;
  if (i < NN) deg[i] = 1.0f;                 // self loop contributes 1
}

__global__ void k_edge_deg(const int* __restrict__ row, float* __restrict__ deg) {
  int e = blockIdx.x * blockDim.x + threadIdx.x;
  if (e < NE) atomicAdd(&deg[row[e]], 1.0f);
}

__global__ void k_dinv(float* __restrict__ d) {
  int i = blockIdx.x * blockDim.x + threadIdx.x;
  if (i < NN) d[i] = rsqrtf(d[i]);           // deg >= 1 always
}

// ---------------- edge aggregation (the memory-bound bottleneck) ----------------

// agg[i][:] = x[i][:] * dinv[i]^2   (self-loop term; also zero-initializes agg)
__global__ void k_self_init(const float* __restrict__ x, const float* __restrict__ dinv,
                            float* __restrict__ agg) {
  int idx = blockIdx.x * blockDim.x + threadIdx.x;  // float4 index, 32 per node
  if (idx < NN * 32) {
    int node = idx >> 5;
    float w = dinv[node]; w = w * w;
    float4 v = ((const float4*)x)[idx];
    v.x *= w; v.y *= w; v.z *= w; v.w *= w;
    ((float4*)agg)[idx] = v;
  }
}

// one wave (32 lanes) per edge: lane loads float4 of x[col], scatter-adds to agg[row]
__global__ void k_edge_agg(const int* __restrict__ row, const int* __restrict__ col,
                           const float* __restrict__ dinv, const float* __restrict__ x,
                           float* __restrict__ agg) {
  int gid  = blockIdx.x * blockDim.x + threadIdx.x;
  int e    = gid >> 5;
  int lane = gid & 31;
  if (e < NE) {
    int r = row[e], c = col[e];
    float w = dinv[r] * dinv[c];
    float4 v = ((const float4*)x)[c * 32 + lane];
    float* dst = agg + (size_t)r * DD + lane * 4;
    atomicAdd(dst + 0, v.x * w);
    atomicAdd(dst + 1, v.y * w);
    atomicAdd(dst + 2, v.z * w);
    atomicAdd(dst + 3, v.w * w);
  }
}

// ---------------- WMMA GEMM: C = [relu](A[rows x K] * W[K x NOUT] + bias) ----------------
// Fully static shapes: per-lane base pointers + immediate offsets, no divergent
// predication around loads (final 40-col layer uses index clamping; a clamped
// B column only feeds the matching D column, which is masked on store).
// Block = WPB waves; blockIdx.x -> 16-row A tile (staged once in LDS);
// each wave computes one 16x16 tile via V_WMMA_F32_16X16X4_F32 (exact fp32).

template<int K, int NOUT, int WPB, bool RELU>
__global__ __launch_bounds__(WPB * 32) void k_gemm(const float* __restrict__ A,
                                                   const float* __restrict__ W,
                                                   const float* __restrict__ bias,
                                                   float* __restrict__ C) {
  constexpr int NTILES = (NOUT + 15) / 16;
  constexpr bool MASK  = (NOUT % 16) != 0;

  __shared__ __align__(16) float As[16 * K];
  const int rowBase = blockIdx.x * 16;

  // cooperative coalesced load of the 16 x K A-tile
  const float4* A4  = (const float4*)(A + (size_t)rowBase * K);
  float4*       As4 = (float4*)As;
  for (int idx = threadIdx.x; idx < 4 * K; idx += WPB * 32) As4[idx] = A4[idx];
  __syncthreads();

  const int wave  = threadIdx.x >> 5;
  const int lane  = threadIdx.x & 31;
  const int lm    = lane & 15;                  // M (A) / N (B,C,D) within tile
  const int koff  = (lane >> 4) << 1;           // K pair base: 0 or 2
  const int nTile = blockIdx.y * WPB + wave;
  if (NTILES % WPB != 0) { if (nTile >= NTILES) return; }  // wave-uniform
  const int n   = nTile * 16 + lm;
  const int ncl = MASK ? ((n < NOUT) ? n : NOUT - 1) : n;  // safe column

  const float* Ap = As + lm * K + koff;                    // LDS base (b64 pairs)
  const float* Wp = W + (size_t)koff * NOUT + ncl;         // global base (imm offs)

  v8f acc = {};
  #pragma unroll
  for (int k = 0; k < K; k += 4) {
    v2f a, b;
    a.x = Ap[k];                    // A[M=lm][k+koff]
    a.y = Ap[k + 1];                // A[M=lm][k+koff+1]
    b.x = Wp[(size_t)k * NOUT];             // B[K=k+koff][N]
    b.y = Wp[(size_t)(k + 1) * NOUT];       // B[K=k+koff+1][N]
    acc = __builtin_amdgcn_wmma_f32_16x16x4_f32(false, a, false, b,
                                                (short)0, acc, false, false);
  }

  // D layout: VGPR g -> M = g (lanes 0-15) / g+8 (lanes 16-31), N = lane%16
  const float bv   = bias[ncl];
  const int   mAdd = (lane >> 4) << 3;
  float* Cp = C + (size_t)(rowBase + mAdd) * NOUT + n;
  #pragma unroll
  for (int g = 0; g < 8; ++g) {
    float v = acc[g] + bv;
    if (RELU) v = fmaxf(v, 0.0f);
    if (!MASK || (n < NOUT)) Cp[(size_t)g * NOUT] = v;
  }
}

// ---------------- launcher ----------------

extern "C" void kernel_launch(void* const* d_in, const int* in_sizes, int n_in,
                              void* d_out, int out_size, void* d_ws, size_t ws_size,
                              hipStream_t stream) {
  const float* x   = (const float*)d_in[0];
  const int*   ei  = (const int*)d_in[1];
  const int*   row = ei;
  const int*   col = ei + NE;
  const float* W0  = (const float*)d_in[2];
  const float* b0  = (const float*)d_in[3];
  const float* W1  = (const float*)d_in[4];
  const float* b1  = (const float*)d_in[5];
  const float* W2  = (const float*)d_in[6];
  const float* b2  = (const float*)d_in[7];
  const float* Wm1 = (const float*)d_in[8];
  const float* bm1 = (const float*)d_in[9];
  const float* Wm2 = (const float*)d_in[10];
  const float* bm2 = (const float*)d_in[11];
  float* out = (float*)d_out;

  // workspace layout (16B aligned): dinv[N], agg[N*128], h[N*128], h2[N*256]
  char*  ws   = (char*)d_ws;
  float* dinv = (float*)ws;
  float* agg  = (float*)(ws + (size_t)NN * 4);
  float* h    = agg + (size_t)NN * DD;
  float* h2   = h   + (size_t)NN * DD;

  const int rowTiles = NN / 16;  // 6250 exactly

  // symmetric GCN normalization
  k_init_deg<<<(NN + 255) / 256, 256, 0, stream>>>(dinv);
  k_edge_deg<<<(NE + 255) / 256, 256, 0, stream>>>(row, dinv);
  k_dinv    <<<(NN + 255) / 256, 256, 0, stream>>>(dinv);

  const float* Ws[3] = {W0, W1, W2};
  const float* bs[3] = {b0, b1, b2};
  for (int l = 0; l < 3; ++l) {
    const float* in = (l == 0) ? x : h;
    k_self_init<<<(NN * 32 + 255) / 256, 256, 0, stream>>>(in, dinv, agg);
    k_edge_agg <<<(NE * 32 + 255) / 256, 256, 0, stream>>>(row, col, dinv, in, agg);
    k_gemm<DD, DD, 8, true><<<dim3(rowTiles, 1), 256, 0, stream>>>(agg, Ws[l], bs[l], h);
  }

  // MLP: relu(h @ Wm1 + bm1) -> h2 ; h2 @ Wm2 + bm2 -> out (40 cols, masked)
  k_gemm<DD, HH, 8, true ><<<dim3(rowTiles, 2), 256, 0, stream>>>(h,  Wm1, bm1, h2);
  k_gemm<HH, LL, 4, false><<<dim3(rowTiles, 1), 128, 0, stream>>>(h2, Wm2, bm2, out);
}